// ConsistencyMetric_24670292148305
// MI455X (gfx1250) — compile-verified
//
#include <hip/hip_runtime.h>
#include <math.h>

typedef __attribute__((ext_vector_type(2))) float v2f;
typedef __attribute__((ext_vector_type(8))) float v8f;

#define B_      32
#define K_      16
#define T_      30
#define N_      128
#define NSM     20
#define NN      (N_ * NSM)    // 2560 nodes per batch
#define KT      (K_ * T_)     // 480 pred points per batch
#define NODE_TILES (NN / 16)  // 160
#define KT_TILES   (KT / 16)  // 30

// v_med3_num_f32 via inline asm: prevents clang from folding med3(x,0,inf)
// back into maxnum/minnum (which lowers with an extra canonicalize op).
__device__ __forceinline__ float med3_asm(float a, float b, float c) {
    float d;
    asm("v_med3_num_f32 %0, %1, %2, %3" : "=v"(d) : "v"(a), "v"(b), "v"(c));
    return d;
}

__global__ __launch_bounds__(256)
void ConsistencyMetric_kernel(const float* __restrict__ preds,
                              const float* __restrict__ cnodes,
                              const int*   __restrict__ cmask,
                              float*       __restrict__ out)
{
    __shared__ float4 nodes[NN];   // {-2x, -2y, |c|^2 or +inf, yaw}  (40 KB)
    __shared__ float  cons[KT];    // per-point min cost               (1.9 KB)

    const int  b    = blockIdx.x;
    const int  tid  = threadIdx.x;
    const int  lane = tid & 31;
    const int  col  = lane & 15;          // tile column owned by this lane
    const bool hi   = (lane & 16) != 0;   // upper half -> K{2,3} / rows 8..15
    const int  wave = tid >> 5;

    // ---------------- Phase 1: build node table in LDS ----------------
    const float2* cnB = (const float2*)cnodes + (size_t)b * NN;
    const int*    mkB = cmask + (size_t)b * NN;
    for (int q = tid; q < NN; q += 256) {
        int    i  = q % NSM;
        float2 p  = cnB[q];
        float2 pm = cnB[(i > 0) ? (q - 1) : q];
        float yaw = (i > 0) ? -atan2f(p.x - pm.x, p.y - pm.y) : 0.0f;
        float c2  = (mkB[q] == 1) ? __builtin_inff() : (p.x * p.x + p.y * p.y);
        nodes[q]  = make_float4(-2.0f * p.x, -2.0f * p.y, c2, yaw);
    }
    __syncthreads();

    const float PI_F   = 3.14159265358979323846f;
    const float TT_PI  = 2.09439510239319549231f;   // 2*pi/3 = pi - pi/3
    const float INF_F  = __builtin_inff();

    // Constants kept live in VGPRs for the med3 inline asm.
    float vzero = 0.0f, vinf = INF_F, vninf = -INF_F;

    // ---------------- Phase 2: WMMA distance tiles + cost + min -------
    const float2* pB = (const float2*)preds + (size_t)b * KT;

    for (int tile = wave; tile < KT_TILES; tile += 8) {
        int    jg   = tile * 16 + col;            // flat k*T + t
        int    t    = jg % T_;
        float2 pp   = pB[jg];
        float2 pm   = pB[(t > 0) ? (jg - 1) : jg];
        float  pyaw = (t > 0) ? -atan2f(pp.x - pm.x, pp.y - pm.y) : 0.0f;
        float  p2v  = pp.x * pp.x + pp.y * pp.y;

        // A (16x4, preds, LOOP-INVARIANT): row m = [px, py, p2, 1]
        v2f Av;
        Av.x = hi ? p2v  : pp.x;
        Av.y = hi ? 1.0f : pp.y;

        // pred yaw for the 8 rows this lane sees (rows r + 8*hi) — once per KT tile
        float pyr[8];
#pragma unroll
        for (int r = 0; r < 8; ++r)
            pyr[r] = __shfl(pyaw, hi ? (r + 8) : r, 32);

        float mn[8];
#pragma unroll
        for (int r = 0; r < 8; ++r) mn[r] = INF_F;

        for (int nt = 0; nt < NODE_TILES; ++nt) {
            float4 nd = nodes[nt * 16 + col];
            // B (4x16, nodes): col j = [-2cx, -2cy, 1, c2]
            v2f Bv;
            Bv.x = hi ? 1.0f : nd.x;
            Bv.y = hi ? nd.z : nd.y;

            v8f C = {0.f, 0.f, 0.f, 0.f, 0.f, 0.f, 0.f, 0.f};
            // D[m][j] = squared distance pred_m <-> node_j (masked -> +inf)
            v8f D = __builtin_amdgcn_wmma_f32_16x16x4_f32(
                        false, Av, false, Bv, (short)0, C, false, false);

            float nyaw = nd.w;                    // this lane's node yaw
#pragma unroll
            for (int r = 0; r < 8; ++r) {
                float d2c = med3_asm(D[r], vzero, vinf);            // relu clamp
                float dd  = __builtin_amdgcn_sqrtf(d2c);            // raw v_sqrt_f32
                float cd  = med3_asm(dd - 2.0f, vzero, vinf);       // relu(d-2)
                float t1  = fabsf(nyaw - pyr[r]) - PI_F;            // |wrap| = pi - |t1|
                float cy  = med3_asm(TT_PI - fabsf(t1), vzero, vinf);
                mn[r]     = med3_asm(mn[r], cd + cy, vninf);        // running min
            }
        }

        // min over the 16 node-columns within each lane half
#pragma unroll
        for (int s = 8; s >= 1; s >>= 1) {
#pragma unroll
            for (int r = 0; r < 8; ++r)
                mn[r] = med3_asm(mn[r], __shfl_xor(mn[r], s, 32), vninf);
        }
        if (col == 0) {
#pragma unroll
            for (int r = 0; r < 8; ++r)
                cons[tile * 16 + (hi ? 8 : 0) + r] = mn[r];
        }
    }
    __syncthreads();

    // ---------------- Phase 3: deterministic sum over T ----------------
    if (tid < K_) {
        float s = 0.0f;
#pragma unroll
        for (int t = 0; t < T_; ++t) s += cons[tid * T_ + t];
        out[(size_t)b * K_ + tid] = s;
    }
}

extern "C" void kernel_launch(void* const* d_in, const int* in_sizes, int n_in,
                              void* d_out, int out_size, void* d_ws, size_t ws_size,
                              hipStream_t stream) {
    (void)in_sizes; (void)n_in; (void)out_size; (void)d_ws; (void)ws_size;
    const float* preds = (const float*)d_in[0];
    const float* cn    = (const float*)d_in[1];
    const int*   mk    = (const int*)d_in[2];
    float*       out   = (float*)d_out;
    ConsistencyMetric_kernel<<<B_, 256, 0, stream>>>(preds, cn, mk, out);
}